// Sinkhorn_77429670412676
// MI455X (gfx1250) — compile-verified
//
#include <hip/hip_runtime.h>

#define BATCH 32
#define NDIM 1024
#define MDIM 1024
#define EPS 1e-4f

typedef __attribute__((ext_vector_type(2))) float v2f;
typedef __attribute__((ext_vector_type(4))) float v4f;
typedef __attribute__((ext_vector_type(8))) float v8f;

// ---------------------------------------------------------------------------
// init: R = C = 1, T = 0   (count = BATCH*1024)
// ---------------------------------------------------------------------------
__global__ void sk_init(float* __restrict__ R, float* __restrict__ C,
                        float* __restrict__ T, int count) {
    int i = blockIdx.x * blockDim.x + threadIdx.x;
    if (i < count) { R[i] = 1.0f; C[i] = 1.0f; T[i] = 0.0f; }
}

// ---------------------------------------------------------------------------
// Column sums via WMMA:  T[b][m] += sum_{n in chunk} A[b][n][m] * R[b][n]
//
// D(16x16) = Amat(16x4) x Bmat(4x16) + D
//   Amat[i][k] = R[n0+k]  (replicated over i)   -> all D rows identical
//   Bmat[k][j] = A[n0+k][m0+j]                  (natural row-major tile)
// A-operand layout (16x4 f32): lanes 0-15 hold K=0,1 (v0,v1); lanes 16-31 K=2,3
// B-operand layout (4x16 f32): v0 = rows {K=0 | K=2}, v1 = rows {K=1 | K=3}
//
// Grid: BATCH * 8 mGroups(128 cols) * 8 nChunks(128 rows) = 2048 blocks x 256.
// Each of the 8 waves per block owns one 16-column tile; 32 chained WMMAs
// accumulate the 128-row chunk, then lanes 0-15 fold acc[0] into T atomically.
// EXEC is all-ones at every WMMA (no divergence before it).
// ---------------------------------------------------------------------------
__global__ void sk_colsum_wmma(const float* __restrict__ A,
                               const float* __restrict__ R,
                               float* __restrict__ T) {
    int blk    = blockIdx.x;
    int nChunk = blk & 7;
    int mGroup = (blk >> 3) & 7;
    int b      = blk >> 6;

    int wave = threadIdx.x >> 5;
    int lane = threadIdx.x & 31;
    int half = lane >> 4;          // 0: lanes 0-15, 1: lanes 16-31
    int col  = mGroup * 128 + wave * 16 + (lane & 15);

    const float* Ab = A + (size_t)b * NDIM * MDIM;
    const float* Rb = R + b * NDIM;

    v8f acc = {};
    int nbase = nChunk * 128;
#pragma unroll 4
    for (int s = 0; s < 128; s += 4) {
        int row = nbase + s + 2 * half;     // this half-wave covers rows K={2h,2h+1}
        v2f a, bb;
        a.x  = Rb[row];
        a.y  = Rb[row + 1];
        bb.x = Ab[(size_t)row * MDIM + col];
        bb.y = Ab[(size_t)(row + 1) * MDIM + col];
        acc = __builtin_amdgcn_wmma_f32_16x16x4_f32(
            /*neg_a=*/false, a, /*neg_b=*/false, bb,
            /*c_mod=*/(short)0, acc, /*reuse_a=*/false, /*reuse_b=*/false);
    }
    // every D row equals the partial column sum; lanes 0-15 hold cols m0..m0+15
    if (lane < 16) atomicAdd(&T[b * MDIM + col], acc[0]);
}

// ---------------------------------------------------------------------------
// Row sums: T[b][n] = sum_m A[b][n][m] * C[b][m]
// One wave per row; lane-strided float4 loads (fully coalesced, contiguous M),
// then 5-step xor-shuffle reduction (wave32). No atomics needed.
// Grid: BATCH*NDIM/8 = 4096 blocks x 256 (8 waves -> 8 rows per block).
// ---------------------------------------------------------------------------
__global__ void sk_rowsum(const float* __restrict__ A,
                          const float* __restrict__ C,
                          float* __restrict__ T) {
    int wave = threadIdx.x >> 5;
    int lane = threadIdx.x & 31;
    int rowG = blockIdx.x * 8 + wave;          // 0 .. BATCH*NDIM-1
    int b    = rowG >> 10;
    int n    = rowG & (NDIM - 1);

    const float4* Arow = (const float4*)(A + ((size_t)b * NDIM + n) * MDIM);
    const float4* Cb   = (const float4*)(C + b * MDIM);

    float acc = 0.0f;
#pragma unroll
    for (int k = 0; k < 8; ++k) {
        int idx  = lane + 32 * k;              // 256 float4 per row
        float4 a = Arow[idx];
        float4 c = Cb[idx];
        acc += a.x * c.x + a.y * c.y + a.z * c.z + a.w * c.w;
    }
#pragma unroll
    for (int off = 16; off > 0; off >>= 1)
        acc += __shfl_xor(acc, off, 32);
    if (lane == 0) T[b * NDIM + n] = acc;
}

// ---------------------------------------------------------------------------
// vec[i] = vec[i] / (vec[i]*T[i] + eps); T[i] = 0  (re-arm accumulator)
// ---------------------------------------------------------------------------
__global__ void sk_update(float* __restrict__ vec, float* __restrict__ T,
                          int count) {
    int i = blockIdx.x * blockDim.x + threadIdx.x;
    if (i < count) {
        float v = vec[i];
        float t = T[i];
        vec[i] = v / (v * t + EPS);
        T[i]   = 0.0f;
    }
}

// ---------------------------------------------------------------------------
// out[b][n][m] = A[b][n][m] * R[b][n] * C[b][m]   (float4 streaming pass)
//
// Output is write-once / never re-read -> non-temporal (TH_NT) stores so the
// 128 MiB result stream does not evict the L2-resident input mid-pass
// (input + output = 256 MiB > 192 MB L2 under RT policy). Input loads are
// also NT: this is the last read of each line.
// Grid: BATCH*NDIM*MDIM/4/256 = 32768 blocks x 256.
// ---------------------------------------------------------------------------
__global__ void sk_final(const float* __restrict__ A,
                         const float* __restrict__ R,
                         const float* __restrict__ C,
                         float* __restrict__ out) {
    size_t g  = (size_t)blockIdx.x * blockDim.x + threadIdx.x;  // float4 index
    int m4 = (int)(g & (MDIM / 4 - 1));
    int n  = (int)((g >> 8) & (NDIM - 1));
    int b  = (int)(g >> 18);

    float r  = R[b * NDIM + n];
    v4f   cv = ((const v4f*)(C + b * MDIM))[m4];
    v4f   av = __builtin_nontemporal_load(((const v4f*)A) + g);
    v4f   o  = av * r * cv;
    __builtin_nontemporal_store(o, ((v4f*)out) + g);
}

// ---------------------------------------------------------------------------
extern "C" void kernel_launch(void* const* d_in, const int* in_sizes, int n_in,
                              void* d_out, int out_size, void* d_ws, size_t ws_size,
                              hipStream_t stream) {
    const float* s0 = (const float*)d_in[0];
    float* out = (float*)d_out;

    const int VEC = BATCH * NDIM;          // 32768 floats per vector
    float* R = (float*)d_ws;
    float* C = R + VEC;
    float* T = C + VEC;

    // init R=C=1, T=0
    sk_init<<<VEC / 256, 256, 0, stream>>>(R, C, T, VEC);

    for (int i = 0; i < 15; ++i) {
        if ((i & 1) == 0) {
            // column sums over N -> update C
            sk_colsum_wmma<<<BATCH * 8 * 8, 256, 0, stream>>>(s0, R, T);
            sk_update<<<VEC / 256, 256, 0, stream>>>(C, T, VEC);
        } else {
            // row sums over M -> update R
            sk_rowsum<<<BATCH * NDIM / 8, 256, 0, stream>>>(s0, C, T);
            sk_update<<<VEC / 256, 256, 0, stream>>>(R, T, VEC);
        }
    }

    // final rescale pass (NT streaming)
    sk_final<<<(BATCH * NDIM * MDIM / 4) / 256, 256, 0, stream>>>(s0, R, C, out);
}